// GATConv_1554778161246
// MI455X (gfx1250) — compile-verified
//
#include <hip/hip_runtime.h>
#include <math.h>

#define IN_CH   256
#define OUT_CH  128   // HEADS*C
#define HEADS   4
#define CPH     32
#define NEG_SLOPE 0.2f
#define LDS_STRIDE 258  // 16x256 tile padded: (258 mod 64)=2 -> conflict-free rows

typedef float v2f __attribute__((ext_vector_type(2)));
typedef float v8f __attribute__((ext_vector_type(8)));

// order-preserving float <-> uint mapping so unsigned atomicMax == float max
__device__ __forceinline__ unsigned ford(float f) {
    unsigned u = __float_as_uint(f);
    return (u & 0x80000000u) ? ~u : (u | 0x80000000u);
}
__device__ __forceinline__ float funord(unsigned e) {
    return (e & 0x80000000u) ? __uint_as_float(e & 0x7fffffffu)
                             : __uint_as_float(~e);
}
__device__ __forceinline__ float lrelu(float x) { return x >= 0.f ? x : NEG_SLOPE * x; }

// ---------------------------------------------------------------------------
// Kernel 1: x_proj = x @ W via V_WMMA_F32_16X16X4_F32.
// Block = 256 threads = 8 waves. Block handles 16 rows (nodes); wave w handles
// columns [16w, 16w+16). K=256 -> 64 WMMA issues per wave.
// A frag (16x4 f32): lanes 0-15 hold row m, K={k0,k0+1}; lanes 16-31 K={k0+2,k0+3}.
// B frag (4x16 f32): mirrored: VGPR0 rows K={k0 | k0+2}, VGPR1 rows K+1, lane = n.
// C/D (16x16 f32): VGPR i = row i (lanes 0-15) / row i+8 (lanes 16-31), lane = n.
// ---------------------------------------------------------------------------
__global__ __launch_bounds__(256)
void gemm_wmma_f32(const float* __restrict__ x, const float* __restrict__ W,
                   float* __restrict__ xp, int N) {
    __shared__ float lx[16 * LDS_STRIDE];
    const int tid = threadIdx.x;
    const int node_base = blockIdx.x * 16;

    // cooperative load: 16x256 floats = 1024 float4, 4 per thread
    #pragma unroll
    for (int i = 0; i < 4; ++i) {
        int f4 = tid + i * 256;
        int r  = f4 >> 6;          // 64 float4 per row
        int c4 = f4 & 63;
        int node = node_base + r;
        if (node >= N) node = N - 1;   // clamp (tail tile): garbage rows never stored
        float4 v = ((const float4*)(x + (size_t)node * IN_CH))[c4];
        float* dst = &lx[r * LDS_STRIDE + c4 * 4];
        dst[0] = v.x; dst[1] = v.y; dst[2] = v.z; dst[3] = v.w;
    }
    __syncthreads();

    const int lane = tid & 31;
    const int wave = tid >> 5;       // column tile 0..7
    const int nb   = wave * 16;
    const int ml   = lane & 15;      // A: row m   B: col n   C/D: col n
    const int hi   = lane >> 4;

    v8f c = {};
    #pragma unroll 8
    for (int k0 = 0; k0 < IN_CH; k0 += 4) {
        const int ka = k0 + 2 * hi;
        v2f a, b;
        const float* ap = &lx[ml * LDS_STRIDE + ka];   // even addr -> ds_load_b64
        a.x = ap[0];
        a.y = ap[1];
        const float* bp = &W[(size_t)ka * OUT_CH + nb + ml];
        b.x = bp[0];
        b.y = bp[OUT_CH];
        c = __builtin_amdgcn_wmma_f32_16x16x4_f32(false, a, false, b,
                                                  (short)0, c, false, false);
    }
    #pragma unroll
    for (int i = 0; i < 8; ++i) {
        int node = node_base + i + 8 * hi;
        if (node < N) xp[(size_t)node * OUT_CH + nb + ml] = c[i];
    }
}

// ---------------------------------------------------------------------------
// Kernel 2: per-(node,head) attention logits + self-loop max init
// ---------------------------------------------------------------------------
__global__ void node_alpha(const float* __restrict__ xp,
                           const float* __restrict__ att_src,
                           const float* __restrict__ att_dst,
                           float* __restrict__ asrc, float* __restrict__ adst,
                           unsigned* __restrict__ mord, int NH) {
    int t = blockIdx.x * blockDim.x + threadIdx.x;
    if (t >= NH) return;
    int h = t & (HEADS - 1);
    int node = t >> 2;
    const float4* xv = (const float4*)(xp + (size_t)node * OUT_CH + h * CPH);
    const float4* as = (const float4*)(att_src + h * CPH);
    const float4* ad = (const float4*)(att_dst + h * CPH);
    float s1 = 0.f, s2 = 0.f;
    #pragma unroll
    for (int i = 0; i < CPH / 4; ++i) {
        float4 v = xv[i], a = as[i], d = ad[i];
        s1 += v.x * a.x + v.y * a.y + v.z * a.z + v.w * a.w;
        s2 += v.x * d.x + v.y * d.y + v.z * d.z + v.w * d.w;
    }
    asrc[t] = s1;
    adst[t] = s2;
    mord[t] = ford(lrelu(s1 + s2));   // self-loop seeds the segment max
}

// ---------------------------------------------------------------------------
// Kernel 3: edge segment-max (atomicMax on order-preserving uint)
// ---------------------------------------------------------------------------
__global__ void edge_max(const long long* __restrict__ ei,
                         const float* __restrict__ asrc,
                         const float* __restrict__ adst,
                         unsigned* __restrict__ mord, int E) {
    int e = blockIdx.x * blockDim.x + threadIdx.x;
    if (e >= E) return;
    int row = (int)ei[e];
    int col = (int)ei[(size_t)E + e];
    #pragma unroll
    for (int h = 0; h < HEADS; ++h) {
        float v = lrelu(asrc[row * HEADS + h] + adst[col * HEADS + h]);
        atomicMax(&mord[col * HEADS + h], ford(v));
    }
}

// ---------------------------------------------------------------------------
// Kernel 4a: init softmax denominator with self-loop term
// ---------------------------------------------------------------------------
__global__ void node_sum_init(const float* __restrict__ asrc,
                              const float* __restrict__ adst,
                              const unsigned* __restrict__ mord,
                              float* __restrict__ ssum, int NH) {
    int t = blockIdx.x * blockDim.x + threadIdx.x;
    if (t >= NH) return;
    float e = lrelu(asrc[t] + adst[t]);
    ssum[t] = expf(e - funord(mord[t]));
}

// ---------------------------------------------------------------------------
// Kernel 4b: edge segment-sum of exp
// ---------------------------------------------------------------------------
__global__ void edge_sum(const long long* __restrict__ ei,
                         const float* __restrict__ asrc,
                         const float* __restrict__ adst,
                         const unsigned* __restrict__ mord,
                         float* __restrict__ ssum, int E) {
    int e = blockIdx.x * blockDim.x + threadIdx.x;
    if (e >= E) return;
    int row = (int)ei[e];
    int col = (int)ei[(size_t)E + e];
    #pragma unroll
    for (int h = 0; h < HEADS; ++h) {
        int ch = col * HEADS + h;
        float v = lrelu(asrc[row * HEADS + h] + adst[ch]);
        atomicAdd(&ssum[ch], expf(v - funord(mord[ch])));
    }
}

// ---------------------------------------------------------------------------
// Kernel 5a: initialize output with self-loop message + bias (fully writes d_out)
// ---------------------------------------------------------------------------
__global__ void out_init(const float* __restrict__ xp,
                         const float* __restrict__ asrc,
                         const float* __restrict__ adst,
                         const unsigned* __restrict__ mord,
                         const float* __restrict__ ssum,
                         const float* __restrict__ bias,
                         float* __restrict__ out, long long total) {
    long long t = (long long)blockIdx.x * blockDim.x + threadIdx.x;
    if (t >= total) return;
    int cidx = (int)(t & (OUT_CH - 1));
    long long node = t >> 7;
    int nh = (int)(node * HEADS) + (cidx >> 5);
    float e = lrelu(asrc[nh] + adst[nh]);
    float alpha = expf(e - funord(mord[nh])) / (ssum[nh] + 1e-16f);
    out[t] = xp[t] * alpha + bias[cidx];
}

// ---------------------------------------------------------------------------
// Kernel 5b: weighted message scatter-add. 128 threads per edge (2 edges/block),
// coalesced 128-wide gather of x_proj[row], f32 atomics into out[col].
// ---------------------------------------------------------------------------
__global__ __launch_bounds__(256)
void edge_scatter(const long long* __restrict__ ei,
                  const float* __restrict__ xp,
                  const float* __restrict__ asrc,
                  const float* __restrict__ adst,
                  const unsigned* __restrict__ mord,
                  const float* __restrict__ ssum,
                  float* __restrict__ out, int E) {
    int e = blockIdx.x * 2 + (threadIdx.x >> 7);
    if (e >= E) return;
    int cidx = threadIdx.x & (OUT_CH - 1);
    int h = cidx >> 5;
    int row = (int)ei[e];
    int col = (int)ei[(size_t)E + e];
    int ch = col * HEADS + h;
    float v = lrelu(asrc[row * HEADS + h] + adst[ch]);
    float alpha = expf(v - funord(mord[ch])) / (ssum[ch] + 1e-16f);
    atomicAdd(&out[(size_t)col * OUT_CH + cidx],
              xp[(size_t)row * OUT_CH + cidx] * alpha);
}

// ---------------------------------------------------------------------------
extern "C" void kernel_launch(void* const* d_in, const int* in_sizes, int n_in,
                              void* d_out, int out_size, void* d_ws, size_t ws_size,
                              hipStream_t stream) {
    const float*     x       = (const float*)d_in[0];
    const long long* ei      = (const long long*)d_in[1];   // int64 [2,E]
    const float*     W       = (const float*)d_in[2];
    const float*     att_src = (const float*)d_in[3];
    const float*     att_dst = (const float*)d_in[4];
    const float*     bias    = (const float*)d_in[5];
    float*           out     = (float*)d_out;

    const int N = in_sizes[0] / IN_CH;
    const int E = in_sizes[1] / 2;
    const int NH = N * HEADS;

    // workspace layout
    float*    xp   = (float*)d_ws;                         // N*128
    float*    asrc = xp + (size_t)N * OUT_CH;              // N*4
    float*    adst = asrc + (size_t)NH;                    // N*4
    unsigned* mord = (unsigned*)(adst + (size_t)NH);       // N*4
    float*    ssum = (float*)(mord + (size_t)NH);          // N*4

    const int row_tiles = (N + 15) / 16;
    gemm_wmma_f32<<<row_tiles, 256, 0, stream>>>(x, W, xp, N);

    node_alpha<<<(NH + 255) / 256, 256, 0, stream>>>(xp, att_src, att_dst,
                                                     asrc, adst, mord, NH);
    edge_max<<<(E + 255) / 256, 256, 0, stream>>>(ei, asrc, adst, mord, E);
    node_sum_init<<<(NH + 255) / 256, 256, 0, stream>>>(asrc, adst, mord, ssum, NH);
    edge_sum<<<(E + 255) / 256, 256, 0, stream>>>(ei, asrc, adst, mord, ssum, E);

    long long total = (long long)N * OUT_CH;
    out_init<<<(unsigned)((total + 255) / 256), 256, 0, stream>>>(
        xp, asrc, adst, mord, ssum, bias, out, total);
    edge_scatter<<<(E + 1) / 2, 256, 0, stream>>>(ei, xp, asrc, adst, mord, ssum,
                                                  out, E);
}